// ModelNew_17411797418166
// MI455X (gfx1250) — compile-verified
//
#include <hip/hip_runtime.h>
#include <stdint.h>

// Reference: out = input.at[indices].set(update)
//   input  : (8192, 64, 128) f32   -> D0 rows of ROW_ELEMS floats
//   indices: (2048,) i32 (unique)
//   update : (2048, 64, 128) f32
// Pure bandwidth problem (0 FLOPs). Strategy: build a row->update-slot flag
// map, then write every output row exactly once, streaming each 32KB row
// through LDS with CDNA5 async global<->LDS DMA (ASYNCcnt path), with
// non-temporal cache hints since every byte is touched exactly once.

#define ROW_ELEMS 8192                    // 64*128 floats per row
#define ROW_BYTES (ROW_ELEMS * 4)         // 32 KB
#define BLOCK 256
#define CHUNK_BYTES (BLOCK * 16)          // 4 KB per sweep (B128 per lane)
#define CHUNKS (ROW_BYTES / CHUNK_BYTES)  // 8

typedef __attribute__((address_space(3))) char lds_char_t;

// Copy one ROW_BYTES row src->dst, staged through LDS via async DMA.
// Each lane loads/stores only its own 16B slices; per-wave ASYNCcnt ordering
// (loads complete in order; wait 0 before stores) makes this race-free
// without any workgroup barrier.
__device__ __forceinline__ void async_row_copy(const char* __restrict__ src,
                                               char* __restrict__ dst,
                                               unsigned ldsbase, unsigned tb) {
#pragma unroll
  for (int i = 0; i < CHUNKS; ++i) {
    asm volatile("global_load_async_to_lds_b128 %0, %1, off th:TH_LOAD_NT"
                 :
                 : "v"(ldsbase + tb + (unsigned)(i * CHUNK_BYTES)),
                   "v"(src + tb + i * CHUNK_BYTES)
                 : "memory");
  }
  asm volatile("s_wait_asynccnt 0" ::: "memory");
#pragma unroll
  for (int i = 0; i < CHUNKS; ++i) {
    asm volatile("global_store_async_from_lds_b128 %0, %1, off th:TH_STORE_NT"
                 :
                 : "v"(dst + tb + i * CHUNK_BYTES),
                   "v"(ldsbase + tb + (unsigned)(i * CHUNK_BYTES))
                 : "memory");
  }
  // s_endpgm performs an implicit wait-idle; stores drain before kernel end.
}

__global__ void zero_flags_kernel(int* __restrict__ flags, int n) {
  int i = blockIdx.x * blockDim.x + threadIdx.x;
  if (i < n) flags[i] = 0;
}

__global__ void scatter_flags_kernel(const int* __restrict__ idx,
                                     int* __restrict__ flags, int k) {
  int i = blockIdx.x * blockDim.x + threadIdx.x;
  if (i < k) flags[idx[i]] = i + 1;  // 0 = "keep input row"
}

// One workgroup per output row: gather from update (if flagged) or input.
__global__ __launch_bounds__(BLOCK) void row_copy_flagged_kernel(
    const float* __restrict__ input, const float* __restrict__ update,
    const int* __restrict__ flags, float* __restrict__ out) {
  __shared__ __align__(16) char lds[ROW_BYTES];
  const unsigned ldsbase = (unsigned)(uintptr_t)(lds_char_t*)lds;
  const unsigned tb = threadIdx.x * 16u;

  const size_t row = blockIdx.x;
  const int f = flags[row];
  const char* src = (f > 0)
                        ? (const char*)(update + (size_t)(f - 1) * ROW_ELEMS)
                        : (const char*)(input + row * ROW_ELEMS);
  char* dst = (char*)(out + row * ROW_ELEMS);
  async_row_copy(src, dst, ldsbase, tb);
}

// Fallback path (workspace too small): full copy, then overwrite-scatter.
__global__ __launch_bounds__(BLOCK) void copy_all_kernel(
    const float* __restrict__ input, float* __restrict__ out) {
  __shared__ __align__(16) char lds[ROW_BYTES];
  const unsigned ldsbase = (unsigned)(uintptr_t)(lds_char_t*)lds;
  const unsigned tb = threadIdx.x * 16u;
  const size_t row = blockIdx.x;
  async_row_copy((const char*)(input + row * ROW_ELEMS),
                 (char*)(out + row * ROW_ELEMS), ldsbase, tb);
}

__global__ __launch_bounds__(BLOCK) void scatter_rows_kernel(
    const float* __restrict__ update, const int* __restrict__ idx,
    float* __restrict__ out) {
  __shared__ __align__(16) char lds[ROW_BYTES];
  const unsigned ldsbase = (unsigned)(uintptr_t)(lds_char_t*)lds;
  const unsigned tb = threadIdx.x * 16u;
  const size_t k = blockIdx.x;
  const size_t r = (size_t)idx[k];
  async_row_copy((const char*)(update + k * ROW_ELEMS),
                 (char*)(out + r * ROW_ELEMS), ldsbase, tb);
}

extern "C" void kernel_launch(void* const* d_in, const int* in_sizes, int n_in,
                              void* d_out, int out_size, void* d_ws,
                              size_t ws_size, hipStream_t stream) {
  const float* input = (const float*)d_in[0];
  const int* indices = (const int*)d_in[1];
  const float* update = (const float*)d_in[2];
  float* out = (float*)d_out;

  const int D0 = in_sizes[0] / ROW_ELEMS;  // 8192
  const int K = in_sizes[1];               // 2048

  if (ws_size >= (size_t)D0 * sizeof(int)) {
    int* flags = (int*)d_ws;
    zero_flags_kernel<<<(D0 + 255) / 256, 256, 0, stream>>>(flags, D0);
    scatter_flags_kernel<<<(K + 255) / 256, 256, 0, stream>>>(indices, flags, K);
    row_copy_flagged_kernel<<<D0, BLOCK, 0, stream>>>(input, update, flags, out);
  } else {
    copy_all_kernel<<<D0, BLOCK, 0, stream>>>(input, out);
    scatter_rows_kernel<<<K, BLOCK, 0, stream>>>(update, indices, out);
  }
}